// CenterLoss_20263655702582
// MI455X (gfx1250) — compile-verified
//
#include <hip/hip_runtime.h>

// CenterLoss forward for MI455X (gfx1250, wave32).
//
// dist_i = || x_i - centers[labels_i] ||^2, clipped to [1e-12, 1e12], mean over B.
// Memory-bound: 128 MB of x streamed once (NT hint), centers (6 MB) stay hot in L2.
// Reduction arithmetic done with V_WMMA_F32_16X16X4_F32: A and B fed the SAME
// registers => accumulator = T*T^T, diagonal = per-row sum of squares (exact f32).

typedef __attribute__((ext_vector_type(2))) float v2f;
typedef __attribute__((ext_vector_type(4))) float v4f;
typedef __attribute__((ext_vector_type(8))) float v8f;

#define BATCH   16384
#define FEATD   2048
#define RPB     16            // rows per block (one 16-row WMMA tile)
#define BK      128           // feature columns per stage
#define NSTAGE  (FEATD / BK)  // 16
#define LSTRIDE 132           // LDS row stride in dwords: %64==4 -> conflict-free A reads,
                              // %4==0 -> b128-aligned stores

__global__ __launch_bounds__(256) void center_loss_main_kernel(
    const float* __restrict__ x,
    const int*   __restrict__ labels,
    const float* __restrict__ centers,
    float*       __restrict__ partial)   // one partial sum per block
{
    __shared__ __align__(16) float sDiff[RPB * LSTRIDE];  // 8448 B staged diff tile
    __shared__ int   sLabel[RPB];
    __shared__ float sPart[8][RPB];                       // per-wave diagonals
    __shared__ float sRow[RPB];

    const int t    = threadIdx.x;
    const int w    = t >> 5;          // wave id 0..7
    const int lane = t & 31;
    const int m    = lane & 15;       // tile row this lane owns in A/B layout
    const int kh   = lane >> 4;       // K half-select (0: K+0..1, 1: K+2..3)
    const int r0   = blockIdx.x * RPB;

    if (t < RPB) sLabel[t] = labels[r0 + t];
    __syncthreads();

    // Hoist loop-invariant row base pointers (kills per-stage LDS label reload +
    // 64-bit address rebuild that sat on the critical path after the barrier).
    const float* __restrict__ xrow0 = x + (size_t)(r0 + w)     * FEATD;
    const float* __restrict__ xrow1 = x + (size_t)(r0 + w + 8) * FEATD;
    const float* __restrict__ crow0 = centers + (size_t)sLabel[w]     * FEATD;
    const float* __restrict__ crow1 = centers + (size_t)sLabel[w + 8] * FEATD;
    float* __restrict__ sD0 = &sDiff[(w)     * LSTRIDE];
    float* __restrict__ sD1 = &sDiff[(w + 8) * LSTRIDE];

    v8f acc = {0.f, 0.f, 0.f, 0.f, 0.f, 0.f, 0.f, 0.f};

    for (int s = 0; s < NSTAGE; ++s) {
        const int col = s * BK + 4 * lane;

        // ---- Stage: coalesced load of 16x128 x-tile + gathered centers, diff -> LDS.
        // Each wave covers one full contiguous 512B row-chunk (lane -> float4).
        {
            const v4f xv0 = __builtin_nontemporal_load((const v4f*)(xrow0 + col)); // TH:NT
            const v4f cv0 = *(const v4f*)(crow0 + col);
            *(v4f*)(sD0 + 4 * lane) = xv0 - cv0;

            const v4f xv1 = __builtin_nontemporal_load((const v4f*)(xrow1 + col)); // TH:NT
            const v4f cv1 = *(const v4f*)(crow1 + col);
            *(v4f*)(sD1 + 4 * lane) = xv1 - cv1;

            if (s + 1 < NSTAGE) {
                __builtin_prefetch(xrow0 + col + BK, 0, 0);
                __builtin_prefetch(xrow1 + col + BK, 0, 0);
            }
        }
        __syncthreads();

        // ---- WMMA: each wave reduces its 16-column K-slice of the stage.
        // Same registers as A and B => acc += T*T^T; diagonal accumulates sum(diff^2).
        const int kc0 = w * 16;
#pragma unroll
        for (int i = 0; i < 4; ++i) {
            const int kc = kc0 + 4 * i;
            const v2f a = *(const v2f*)(&sDiff[m * LSTRIDE + kc + 2 * kh]);
            acc = __builtin_amdgcn_wmma_f32_16x16x4_f32(
                false, a, false, a, (short)0, acc, false, false);
        }
        __syncthreads();
    }

    // ---- Extract accumulator diagonal: lanes 0..7 -> row=lane at acc[lane],
    //      lanes 24..31 -> row=lane-16 at acc[lane-24].
    const int sel = (lane < 8) ? lane : ((lane >= 24) ? (lane - 24) : -1);
    float diag = 0.0f;
#pragma unroll
    for (int j = 0; j < 8; ++j) diag = (sel == j) ? acc[j] : diag;
    if (sel >= 0) {
        const int rowId = (lane < 8) ? lane : (lane - 16);
        sPart[w][rowId] = diag;          // every (wave,row) slot written exactly once
    }
    __syncthreads();

    // ---- Deterministic fixed-order combine: 8 wave-partials per row, clip, block sum.
    if (t < RPB) {
        float d = 0.0f;
#pragma unroll
        for (int ww = 0; ww < 8; ++ww) d += sPart[ww][t];
        sRow[t] = fminf(fmaxf(d, 1e-12f), 1e12f);
    }
    __syncthreads();
    if (t == 0) {
        float bs = 0.0f;
#pragma unroll
        for (int r = 0; r < RPB; ++r) bs += sRow[r];
        partial[blockIdx.x] = bs;
    }
}

// Final deterministic tree reduction of the 1024 block partials -> mean.
__global__ __launch_bounds__(256) void center_loss_reduce_kernel(
    const float* __restrict__ partial, float* __restrict__ out)
{
    __shared__ float sdata[256];
    const int t = threadIdx.x;
    sdata[t] = partial[t] + partial[t + 256] + partial[t + 512] + partial[t + 768];
    __syncthreads();
#pragma unroll
    for (int stride = 128; stride > 0; stride >>= 1) {
        if (t < stride) sdata[t] += sdata[t + stride];
        __syncthreads();
    }
    if (t == 0) *out = sdata[0] * (1.0f / (float)BATCH);
}

extern "C" void kernel_launch(void* const* d_in, const int* in_sizes, int n_in,
                              void* d_out, int out_size, void* d_ws, size_t ws_size,
                              hipStream_t stream) {
    (void)in_sizes; (void)n_in; (void)out_size; (void)ws_size;
    const float* x       = (const float*)d_in[0];
    const int*   labels  = (const int*)d_in[1];
    const float* centers = (const float*)d_in[2];
    float*       out     = (float*)d_out;
    float*       partial = (float*)d_ws;   // 1024 floats of scratch

    center_loss_main_kernel<<<BATCH / RPB, 256, 0, stream>>>(x, labels, centers, partial);
    center_loss_reduce_kernel<<<1, 256, 0, stream>>>(partial, out);
}